// Retinanet_8521215115411
// MI455X (gfx1250) — compile-verified
//
#include <hip/hip_runtime.h>
#include <cstdint>

// ---------------------------------------------------------------------------
// RetinaNet head on MI455X (gfx1250): conv3x3 as implicit GEMM on the fp32
// WMMA pipe (V_WMMA_F32_16X16X4_F32). M = out-channels, N = N*H*W pixels,
// K = FEAT*9 = 2304.
//
// Block tile: 128 O x 64 P, 256 threads (8 wave32s). Each wave owns one
// 16-row O strip x 64 pixels = 4 accumulators, so one A fragment feeds 4
// WMMAs. K consumed 32-deep per chunk (9 taps x 8 chunks = 72). Software
// pipeline: global->regs for chunk k+1 overlaps WMMA of chunk k; regs->LDS
// commit + ONE barrier per chunk, double-buffered LDS.
//
// LDS uses a K-PAIR-INTERLEAVED layout: element (k, col) lives at
// ((k>>1)*STR + col)*2 + (k&1), so each WMMA fragment (k, k+1 @ same col)
// is one aligned ds_load_b64 into an even VGPR pair. Fragment loads use a
// single per-chunk VGPR base + 16-bit immediate DS offsets, and a 2-stage
// fragment pipeline keeps >=5 DS loads in flight under the WMMAs.
// STR == 16 (mod 32) puts the two half-waves on disjoint 32-bank sets.
// ---------------------------------------------------------------------------

typedef float v2f __attribute__((ext_vector_type(2)));
typedef float v8f __attribute__((ext_vector_type(8)));

#define FEAT   256   // input channels for every conv in this head
#define TILE_O 128
#define TILE_P 64
#define KC     32    // K-depth staged per chunk
#define KROWS  (KC / 2)
#define ASTR   144   // pair-row column stride for A (128 cols + pad), 144 % 32 == 16
#define BSTR   80    // pair-row column stride for B ( 64 cols + pad),  80 % 32 == 16

// ACT: 0 = ReLU (towers), 1 = identity (box pred), 2 = sigmoid (cls pred)
template<int ACT>
__global__ __launch_bounds__(256, 2)
void conv3x3_wmma_f32(const float* __restrict__ X,    // [N, FEAT, H, W]
                      const float* __restrict__ Wt,   // [OC, FEAT, 3, 3]
                      const float* __restrict__ Bias, // [OC]
                      float* __restrict__ Y,          // [N, OC, H, W]
                      int OC, int H, int Wd, int P /* N*H*W */)
{
    __shared__ float As[2][KROWS * ASTR * 2]; // weights, k-pair interleaved
    __shared__ float Bs[2][KROWS * BSTR * 2]; // patches, k-pair interleaved

    const int tid  = threadIdx.x;
    const int lane = tid & 31;
    const int wave = tid >> 5;           // 0..7 -> 16-row O strip

    const int o_base = blockIdx.y * TILE_O;
    const int p_base = blockIdx.x * TILE_P;
    const int HW     = H * Wd;

    // ---- A staging map: o = tid>>1 (0..127), 16 consecutive k per thread ----
    const int  sa_o  = tid >> 1;
    const int  sa_i0 = (tid & 1) * 16;   // k base (even)
    const int  sa_go = o_base + sa_o;
    const bool sa_ok = (sa_go < OC);
    const float* wrow = Wt + (size_t)(sa_ok ? sa_go : 0) * FEAT * 9;

    // ---- B staging map: 4 pixels x adjacent k-pair per thread ----
    const int sb_pr = (tid & 15) * 4;    // pixel base, float4-aligned
    const int sb_k0 = (tid >> 4) * 2;    // even k of this thread's pair
    size_t nb[4]; int ph[4], pw[4]; bool pv[4];
    #pragma unroll
    for (int j = 0; j < 4; ++j) {
        const int p = p_base + sb_pr + j;
        pv[j] = (p < P);
        int n = 0, h = 0, w = 0;
        if (pv[j]) { n = p / HW; int r = p - n * HW; h = r / Wd; w = r - h * Wd; }
        nb[j] = (size_t)n * FEAT * HW;
        ph[j] = h; pw[j] = w;
    }

    v8f acc[4];
    #pragma unroll
    for (int t = 0; t < 4; ++t) acc[t] = (v8f){0,0,0,0,0,0,0,0};

    // Per-lane fragment addressing (ISA 32-bit A 16x4 layout; B mirrored).
    const int frag_m = lane & 15;
    const int kmsb   = lane >> 4;        // selects pair-row within a k-step
    const int a_col  = wave * 16 + frag_m;

    const int NCH = 9 * (FEAT / KC);     // 72 chunks

    float aR[16];      // pipelined weight stage
    float bR[2][4];    // pipelined patch stage (k-pair x 4 pixels)

    auto load_chunk = [&](int chunk) {
        const int tap = chunk >> 3;              // FEAT/KC == 8
        const int c0  = (chunk & 7) * KC;
        const int kh  = tap / 3 - 1;
        const int kw  = tap % 3 - 1;
        // weights: 16 strided reads (L2-resident; reused by every pixel block)
        #pragma unroll
        for (int j = 0; j < 16; ++j)
            aR[j] = sa_ok ? wrow[(size_t)(c0 + sa_i0 + j) * 9 + tap] : 0.f;
        if (c0 + KC < FEAT)   // warm next sub-row (global_prefetch_b8)
            __builtin_prefetch(wrow + (size_t)(c0 + sa_i0 + KC) * 9 + tap, 0, 0);
        // shifted input patches (im2col with zero padding)
        #pragma unroll
        for (int ii = 0; ii < 2; ++ii) {
            const int c = c0 + sb_k0 + ii;
            #pragma unroll
            for (int j = 0; j < 4; ++j) {
                const int ih = ph[j] + kh, iw = pw[j] + kw;
                float v = 0.f;
                if (pv[j] && ih >= 0 && ih < H && iw >= 0 && iw < Wd)
                    v = X[nb[j] + (size_t)c * HW + (size_t)ih * Wd + iw];
                bR[ii][j] = v;
            }
        }
    };

    auto commit = [&](int sel) {
        // A: 8 adjacent k-pairs -> 8 ds_store_b64
        #pragma unroll
        for (int m = 0; m < 8; ++m) {
            const int row = (sa_i0 >> 1) + m;
            *reinterpret_cast<v2f*>(&As[sel][(row * ASTR + sa_o) * 2]) =
                (v2f){aR[2 * m], aR[2 * m + 1]};
        }
        // B: (2 pixels x k-pair) per float4 -> 2 ds_store_b128
        const int brow = sb_k0 >> 1;
        #pragma unroll
        for (int j2 = 0; j2 < 2; ++j2)
            *reinterpret_cast<float4*>(&Bs[sel][(brow * BSTR + sb_pr + 2 * j2) * 2]) =
                make_float4(bR[0][2 * j2], bR[1][2 * j2],
                            bR[0][2 * j2 + 1], bR[1][2 * j2 + 1]);
    };

    auto compute = [&](int sel) {
        // One VGPR base per operand; every fragment load below uses a pure
        // compile-time byte offset (fits the 16-bit DS offset field).
        const float* asp = &As[sel][(kmsb * ASTR + a_col) * 2];
        const float* bsp = &Bs[sel][(kmsb * BSTR + frag_m) * 2];

        v2f aC, bC[4], aN, bN[4];
        aC = *reinterpret_cast<const v2f*>(asp);
        #pragma unroll
        for (int t = 0; t < 4; ++t)
            bC[t] = *reinterpret_cast<const v2f*>(bsp + t * 32);

        #pragma unroll
        for (int kk = 0; kk < KC / 4; ++kk) {
            if (kk + 1 < KC / 4) {   // prefetch next k-step's fragments
                aN = *reinterpret_cast<const v2f*>(asp + (kk + 1) * (2 * ASTR * 2));
                #pragma unroll
                for (int t = 0; t < 4; ++t)
                    bN[t] = *reinterpret_cast<const v2f*>(
                                bsp + (kk + 1) * (2 * BSTR * 2) + t * 32);
            }
            #pragma unroll
            for (int t = 0; t < 4; ++t)
                acc[t] = __builtin_amdgcn_wmma_f32_16x16x4_f32(
                             false, aC, false, bC[t], (short)0, acc[t], false, false);
            aC = aN;
            #pragma unroll
            for (int t = 0; t < 4; ++t) bC[t] = bN[t];
        }
    };

    // ---- software-pipelined main loop: 1 barrier per chunk ----
    load_chunk(0);
    int sel = 0;
    for (int ch = 0; ch < NCH; ++ch) {
        commit(sel);                 // regs -> LDS buffer sel
        __syncthreads();             // LDS buffer sel visible; prior compute done
        if (ch + 1 < NCH) load_chunk(ch + 1);   // HBM loads fly under WMMAs
        compute(sel);                // WMMA from LDS buffer sel
        sel ^= 1;
    }

    // ---- epilogue: bias + activation, store NCHW (coalesced over pixels) ----
    #pragma unroll
    for (int t = 0; t < 4; ++t) {
        const int p = p_base + t * 16 + frag_m;
        if (p >= P) continue;
        const int n = p / HW; int r = p - n * HW;
        const int h = r / Wd; const int w = r - h * Wd;
        float* yp = Y + (size_t)n * OC * HW + (size_t)h * Wd + w;
        #pragma unroll
        for (int rr = 0; rr < 8; ++rr) {
            const int o = o_base + wave * 16 + kmsb * 8 + rr;
            if (o >= OC) continue;
            float v = acc[t][rr] + Bias[o];
            if (ACT == 0)      v = fmaxf(v, 0.f);
            else if (ACT == 2) v = 1.f / (1.f + __expf(-v));
            yp[(size_t)o * HW] = v;
        }
    }
}

// ---------------------------------------------------------------------------
// Host: chain 10 convs per FPN level through two ping-pong ws buffers.
// ---------------------------------------------------------------------------

extern "C" void kernel_launch(void* const* d_in, const int* in_sizes, int n_in,
                              void* d_out, int out_size, void* d_ws, size_t ws_size,
                              hipStream_t stream)
{
    (void)in_sizes; (void)n_in; (void)out_size; (void)ws_size;

    const float* x[5];
    for (int k = 0; k < 5; ++k) x[k] = (const float*)d_in[k];
    const float* cls_tw = (const float*)d_in[5];
    const float* cls_tb = (const float*)d_in[6];
    const float* box_tw = (const float*)d_in[7];
    const float* box_tb = (const float*)d_in[8];
    const float* cls_pw = (const float*)d_in[9];
    const float* cls_pb = (const float*)d_in[10];
    const float* box_pw = (const float*)d_in[11];
    const float* box_pb = (const float*)d_in[12];

    static const int HS[5] = {80, 40, 20, 10, 5};
    static const int WS[5] = {80, 40, 20, 10, 5};

    // Two ping-pong activation buffers sized for the largest level.
    const size_t MAXE = (size_t)8 * FEAT * 80 * 80;  // 13.1M floats = 52.4 MB
    float* buf0 = (float*)d_ws;
    float* buf1 = buf0 + MAXE;

    float* out = (float*)d_out;
    size_t clsOff[5], boxOff[5], off = 0;
    for (int k = 0; k < 5; ++k) { clsOff[k] = off; off += (size_t)8 * 729 * HS[k] * WS[k]; }
    for (int k = 0; k < 5; ++k) { boxOff[k] = off; off += (size_t)8 * 36  * HS[k] * WS[k]; }

    const size_t TWS = (size_t)FEAT * FEAT * 9;      // one tower conv's weights

    auto conv = [&](const float* Xp, const float* Wp, const float* Bp, float* Yp,
                    int OC, int H, int W, int act) {
        const int P = 8 * H * W;
        dim3 grid((P + TILE_P - 1) / TILE_P, (OC + TILE_O - 1) / TILE_O);
        dim3 block(256);
        switch (act) {
            case 0: conv3x3_wmma_f32<0><<<grid, block, 0, stream>>>(Xp, Wp, Bp, Yp, OC, H, W, P); break;
            case 1: conv3x3_wmma_f32<1><<<grid, block, 0, stream>>>(Xp, Wp, Bp, Yp, OC, H, W, P); break;
            default: conv3x3_wmma_f32<2><<<grid, block, 0, stream>>>(Xp, Wp, Bp, Yp, OC, H, W, P); break;
        }
    };

    for (int k = 0; k < 5; ++k) {
        const int H = HS[k], W = WS[k];

        // ---- classification tower + pred (sigmoid) ----
        const float* tw = cls_tw + (size_t)k * 4 * TWS;
        const float* tb = cls_tb + (size_t)k * 4 * FEAT;
        conv(x[k], tw,           tb,          buf0, FEAT, H, W, 0);
        conv(buf0, tw + 1 * TWS, tb + FEAT,   buf1, FEAT, H, W, 0);
        conv(buf1, tw + 2 * TWS, tb + 2*FEAT, buf0, FEAT, H, W, 0);
        conv(buf0, tw + 3 * TWS, tb + 3*FEAT, buf1, FEAT, H, W, 0);
        conv(buf1, cls_pw + (size_t)k * 729 * FEAT * 9, cls_pb + (size_t)k * 729,
             out + clsOff[k], 729, H, W, 2);

        // ---- box tower + pred (identity) ----
        const float* vw = box_tw + (size_t)k * 4 * TWS;
        const float* vb = box_tb + (size_t)k * 4 * FEAT;
        conv(x[k], vw,           vb,          buf0, FEAT, H, W, 0);
        conv(buf0, vw + 1 * TWS, vb + FEAT,   buf1, FEAT, H, W, 0);
        conv(buf1, vw + 2 * TWS, vb + 2*FEAT, buf0, FEAT, H, W, 0);
        conv(buf0, vw + 3 * TWS, vb + 3*FEAT, buf1, FEAT, H, W, 0);
        conv(buf1, box_pw + (size_t)k * 36 * FEAT * 9, box_pb + (size_t)k * 36,
             out + boxOff[k], 36, H, W, 1);
    }
}